// DifferentiableTopK_43602507989316
// MI455X (gfx1250) — compile-verified
//
#include <hip/hip_runtime.h>

typedef __attribute__((ext_vector_type(2))) float v2f;
typedef __attribute__((ext_vector_type(8))) float v8f;

static constexpr int Bn = 32;    // batch
static constexpr int Nn = 2048;  // row length
static constexpr int Kk = 256;   // top-k columns

#define INV_EPS 1000.0f
#define LOG2E   1.44269504088896340736f
#define LN2     0.69314718055994530942f
#define NEG_INF (-3.0e38f)

// -------- Kernel 1: stable descending rank-sort (one workgroup per batch row)
__global__ __launch_bounds__(256) void k_sort(const float* __restrict__ S,
                                              float* __restrict__ T) {
  __shared__ float s_sh[Nn];
  const int b = blockIdx.x, tid = threadIdx.x;
  for (int k = tid; k < Nn; k += 256) s_sh[k] = S[b * Nn + k];
  __syncthreads();
  for (int i = tid; i < Nn; i += 256) {
    const float v = s_sh[i];
    int cnt = 0;
    for (int j = 0; j < Nn; ++j) {
      const float sj = s_sh[j];
      cnt += (int)((sj > v) || (sj == v && j < i));  // stable, handles ties
    }
    T[b * Nn + cnt] = v;  // ranks form a permutation -> every slot written
  }
}

// -------- Kernel 2: c1[j] = LSE_i M[i,j], WMMA-tiled cross term.
// Wave owns 16 columns; per 16x16 tile: P = outer(s_tile, t_tile) via
// v_wmma_f32_16x16x4_f32 (A col in K=0, B row in K=0), then
// M = 2000*P - 1000*s_i^2 - 1000*t_j^2 per lane-element (C layout:
// lanes 0-15: M=v,N=lane ; lanes 16-31: M=v+8,N=lane-16).
// LDS reads use (lane&15) so every lane issues an unconditional broadcast
// load; the A/B zeroing for the unused K-slots is a v_cndmask, not EXEC.
__global__ __launch_bounds__(256) void k_c1(const float* __restrict__ S,
                                            const float* __restrict__ T,
                                            float* __restrict__ C1) {
  __shared__ float s_sh[Nn];
  __shared__ float q_sh[Nn];
  const int b = blockIdx.y, tid = threadIdx.x;
  for (int k = tid; k < Nn; k += 256) {
    float v = S[b * Nn + k];
    s_sh[k] = v;
    q_sh[k] = v * v;
  }
  __syncthreads();
  const int lane = tid & 31;
  const int l16 = lane & 15;
  const int j = blockIdx.x * 128 + (tid >> 5) * 16 + l16;
  const float tj = T[b * Nn + j];
  const float cj = -INV_EPS * tj * tj;
  const bool lo = (lane < 16);
  const int off = lo ? 0 : 8;
  v2f bmat;
  bmat.x = lo ? tj : 0.0f;  // B[K=0, N=lane] ; K=1..3 rows zero
  bmat.y = 0.0f;

  float mx = NEG_INF;
  for (int i0 = 0; i0 < Nn; i0 += 16) {
    const float sv = s_sh[i0 + l16];  // broadcast-valid for all 32 lanes
    v2f amat;
    amat.x = lo ? sv : 0.0f;  // A[M=lane, K=0] ; K=1..3 zero (cndmask)
    amat.y = 0.0f;
    v8f acc = {};
    v8f p = __builtin_amdgcn_wmma_f32_16x16x4_f32(false, amat, false, bmat,
                                                  (short)0, acc, false, false);
    const int base = i0 + off;
#pragma unroll
    for (int v = 0; v < 8; ++v) {
      float m = fmaf(2.0f * INV_EPS, p[v], fmaf(-INV_EPS, q_sh[base + v], cj));
      mx = fmaxf(mx, m);
    }
  }
  mx = fmaxf(mx, __shfl_xor(mx, 16, 32));  // merge the two half-columns

  float sum = 0.0f;
  for (int i0 = 0; i0 < Nn; i0 += 16) {
    const float sv = s_sh[i0 + l16];
    v2f amat;
    amat.x = lo ? sv : 0.0f;
    amat.y = 0.0f;
    v8f acc = {};
    v8f p = __builtin_amdgcn_wmma_f32_16x16x4_f32(false, amat, false, bmat,
                                                  (short)0, acc, false, false);
    const int base = i0 + off;
#pragma unroll
    for (int v = 0; v < 8; ++v) {
      float m = fmaf(2.0f * INV_EPS, p[v], fmaf(-INV_EPS, q_sh[base + v], cj));
      sum += __builtin_amdgcn_exp2f((m - mx) * LOG2E);
    }
  }
  sum += __shfl_xor(sum, 16, 32);
  if (lo) C1[b * Nn + j] = mx + __builtin_amdgcn_logf(sum) * LN2;
}

// -------- Kernel 3: r1[i] = LSE_j (M[i,j] - c1[j])   (thread per row i)
__global__ __launch_bounds__(256) void k_r1(const float* __restrict__ S,
                                            const float* __restrict__ T,
                                            const float* __restrict__ C1,
                                            float* __restrict__ R1) {
  __shared__ float t_sh[Nn];
  __shared__ float c_sh[Nn];
  const int b = blockIdx.y, tid = threadIdx.x;
  for (int k = tid; k < Nn; k += 256) {
    t_sh[k] = T[b * Nn + k];
    c_sh[k] = C1[b * Nn + k];
  }
  __syncthreads();
  const int i = blockIdx.x * 256 + tid;
  const float si = S[b * Nn + i];
  float mx = NEG_INF;
  for (int j = 0; j < Nn; ++j) {
    float d = si - t_sh[j];
    float a = fmaf(-INV_EPS * d, d, -c_sh[j]);
    mx = fmaxf(mx, a);
  }
  float sum = 0.0f;
  for (int j = 0; j < Nn; ++j) {
    float d = si - t_sh[j];
    float a = fmaf(-INV_EPS * d, d, -c_sh[j]);
    sum += __builtin_amdgcn_exp2f((a - mx) * LOG2E);
  }
  R1[b * Nn + i] = mx + __builtin_amdgcn_logf(sum) * LN2;
}

// -------- Kernel 4: C[j] = c1[j]+c2[j] = LSE_i (M[i,j] - r1[i])  (c1 cancels)
__global__ __launch_bounds__(256) void k_C(const float* __restrict__ S,
                                           const float* __restrict__ T,
                                           const float* __restrict__ R1,
                                           float* __restrict__ Cc) {
  __shared__ float s_sh[Nn];
  __shared__ float r_sh[Nn];
  const int b = blockIdx.y, tid = threadIdx.x;
  for (int k = tid; k < Nn; k += 256) {
    s_sh[k] = S[b * Nn + k];
    r_sh[k] = R1[b * Nn + k];
  }
  __syncthreads();
  const int j = blockIdx.x * 256 + tid;
  const float tj = T[b * Nn + j];
  float mx = NEG_INF;
  for (int i = 0; i < Nn; ++i) {
    float d = s_sh[i] - tj;
    float a = fmaf(-INV_EPS * d, d, -r_sh[i]);
    mx = fmaxf(mx, a);
  }
  float sum = 0.0f;
  for (int i = 0; i < Nn; ++i) {
    float d = s_sh[i] - tj;
    float a = fmaf(-INV_EPS * d, d, -r_sh[i]);
    sum += __builtin_amdgcn_exp2f((a - mx) * LOG2E);
  }
  Cc[b * Nn + j] = mx + __builtin_amdgcn_logf(sum) * LN2;
}

// -------- Kernel 5: out[i] = LSE_{j<K}(M-C) - LSE_{all j}(M-C)   (R cancels)
__global__ __launch_bounds__(256) void k_out(const float* __restrict__ S,
                                             const float* __restrict__ T,
                                             const float* __restrict__ Cc,
                                             float* __restrict__ O) {
  __shared__ float t_sh[Nn];
  __shared__ float c_sh[Nn];
  const int b = blockIdx.y, tid = threadIdx.x;
  for (int k = tid; k < Nn; k += 256) {
    t_sh[k] = T[b * Nn + k];
    c_sh[k] = Cc[b * Nn + k];
  }
  __syncthreads();
  const int i = blockIdx.x * 256 + tid;
  const float si = S[b * Nn + i];
  float mxK = NEG_INF;
  for (int j = 0; j < Kk; ++j) {
    float d = si - t_sh[j];
    float a = fmaf(-INV_EPS * d, d, -c_sh[j]);
    mxK = fmaxf(mxK, a);
  }
  float mxA = mxK;
  for (int j = Kk; j < Nn; ++j) {
    float d = si - t_sh[j];
    float a = fmaf(-INV_EPS * d, d, -c_sh[j]);
    mxA = fmaxf(mxA, a);
  }
  float sK = 0.0f, sA = 0.0f;
  for (int j = 0; j < Kk; ++j) {
    float d = si - t_sh[j];
    float a = fmaf(-INV_EPS * d, d, -c_sh[j]);
    sK += __builtin_amdgcn_exp2f((a - mxK) * LOG2E);
    sA += __builtin_amdgcn_exp2f((a - mxA) * LOG2E);
  }
  for (int j = Kk; j < Nn; ++j) {
    float d = si - t_sh[j];
    float a = fmaf(-INV_EPS * d, d, -c_sh[j]);
    sA += __builtin_amdgcn_exp2f((a - mxA) * LOG2E);
  }
  O[b * Nn + i] = (mxK + __builtin_amdgcn_logf(sK) * LN2) -
                  (mxA + __builtin_amdgcn_logf(sA) * LN2);
}

extern "C" void kernel_launch(void* const* d_in, const int* in_sizes, int n_in,
                              void* d_out, int out_size, void* d_ws,
                              size_t ws_size, hipStream_t stream) {
  (void)in_sizes; (void)n_in; (void)out_size; (void)ws_size;
  const float* scores = (const float*)d_in[0];
  float* out = (float*)d_out;
  float* ws = (float*)d_ws;
  float* t  = ws;                 // [B,N] sorted-descending scores
  float* c1 = ws + 1 * Bn * Nn;   // [B,N]
  float* r1 = ws + 2 * Bn * Nn;   // [B,N]
  float* Cm = ws + 3 * Bn * Nn;   // [B,N] combined column offsets

  k_sort<<<dim3(Bn), dim3(256), 0, stream>>>(scores, t);
  k_c1 <<<dim3(Nn / 128, Bn), dim3(256), 0, stream>>>(scores, t, c1);
  k_r1 <<<dim3(Nn / 256, Bn), dim3(256), 0, stream>>>(scores, t, c1, r1);
  k_C  <<<dim3(Nn / 256, Bn), dim3(256), 0, stream>>>(scores, t, r1, Cm);
  k_out<<<dim3(Nn / 256, Bn), dim3(256), 0, stream>>>(scores, t, Cm, out);
}